// OverlapEntropy_10316511445039
// MI455X (gfx1250) — compile-verified
//
#include <hip/hip_runtime.h>

typedef __attribute__((ext_vector_type(2))) float v2f;
typedef __attribute__((ext_vector_type(8))) float v8f;

#define B_ 4
#define C_ 64
#define H_ 256
#define W_ 256
#define HW_ (H_ * W_)

// Softmax entropy over a 3x3 zero-padded window using
// H = ln(S) - (sum e_i * d_i) / S, d_i = x_i - max  (9 exp + 1 log + 1 rcp).
// Out-of-bounds taps are 0.0 (matches nn.Unfold zero padding, which the
// reference includes in the softmax). Addresses are clamped so loads are
// always in-bounds; masking is v_cndmask (no divergence -> EXEC stays full
// for the WMMA accumulate).
__device__ __forceinline__ float entropy3x3(const float* __restrict__ p,
                                            int om, int o0, int op,
                                            int wm, int w0, int wp,
                                            bool t, bool bo, bool l, bool r) {
    float x0 = p[om + wm], x1 = p[om + w0], x2 = p[om + wp];
    float x3 = p[o0 + wm], x4 = p[o0 + w0], x5 = p[o0 + wp];
    float x6 = p[op + wm], x7 = p[op + w0], x8 = p[op + wp];
    float v[9];
    v[0] = (t && l) ? x0 : 0.f;
    v[1] = t        ? x1 : 0.f;
    v[2] = (t && r) ? x2 : 0.f;
    v[3] = l        ? x3 : 0.f;
    v[4] = x4;
    v[5] = r        ? x5 : 0.f;
    v[6] = (bo && l) ? x6 : 0.f;
    v[7] = bo        ? x7 : 0.f;
    v[8] = (bo && r) ? x8 : 0.f;

    float mx = v[0];
#pragma unroll
    for (int j = 1; j < 9; ++j) mx = fmaxf(mx, v[j]);
    float S = 0.f, dot = 0.f;
#pragma unroll
    for (int j = 0; j < 9; ++j) {
        float d = v[j] - mx;
        float e = __expf(d);
        S += e;
        dot = __fmaf_rn(e, d, dot);
    }
    float Hc = __logf(S) - dot * __builtin_amdgcn_rcpf(S);
    return fmaxf(Hc, 0.f);   // entropy >= 0; keeps uint-punned atomics valid
}

__global__ void entropy_init_minmax(unsigned* __restrict__ mm) {
    int t = threadIdx.x;
    if (t < 4)       mm[t] = 0x7F800000u;  // +inf  (min seed)
    else if (t < 8)  mm[t] = 0x00000000u;  // 0.0   (max seed)
}

// Block = 256 threads = 8 waves. Each wave owns 16 consecutive pixels of one
// row. Lane L computes per-channel entropies for pixel L%16, channels
// 4i + 2*(L/16) + {0,1}, feeding V_WMMA_F32_16X16X4_F32 with an all-ones B so
// the accumulator column n holds the channel sum for pixel row m.
__global__ void __launch_bounds__(256)
entropy_map_kernel(const float* __restrict__ x, float* __restrict__ out,
                   unsigned* __restrict__ mm) {
    __shared__ float sMin[8];
    __shared__ float sMax[8];

    const int tid  = threadIdx.x;
    const int lane = tid & 31;
    const int wave = tid >> 5;

    const int tile = blockIdx.x;          // B * H * (W/128) = 2048 blocks
    const int wt   = tile & 1;            // which 128-wide half of the row
    const int h    = (tile >> 1) & (H_ - 1);
    const int b    = tile >> 9;

    const int px = lane & 15;             // pixel (row M of A)
    const int kh = lane >> 4;             // 0: K=0,1   1: K=2,3
    const int w  = wt * 128 + wave * 16 + px;

    const bool t  = (h > 0);
    const bool bo = (h < H_ - 1);
    const bool l  = (w > 0);
    const bool r  = (w < W_ - 1);
    const int om = (t ? h - 1 : 0) * W_;
    const int o0 = h * W_;
    const int op = (bo ? h + 1 : h) * W_;
    const int wm = l ? w - 1 : 0;
    const int wp = r ? w + 1 : w;

    const float* xb = x + (size_t)b * C_ * HW_;

    v8f acc = {0.f, 0.f, 0.f, 0.f, 0.f, 0.f, 0.f, 0.f};
    const v2f ones = {1.f, 1.f};

#pragma unroll 1
    for (int i = 0; i < 16; ++i) {
        const int ch = 4 * i + 2 * kh;
        const float* xc0 = xb + (size_t)ch * HW_;
        const float* xc1 = xc0 + HW_;
        v2f a;
        a.x = entropy3x3(xc0, om, o0, op, wm, w, wp, t, bo, l, r);
        a.y = entropy3x3(xc1, om, o0, op, wm, w, wp, t, bo, l, r);
        // D = A(16x4) * ones(4x16) + C : channel-group sum per pixel row
        acc = __builtin_amdgcn_wmma_f32_16x16x4_f32(
            false, a, false, ones, (short)0, acc, false, false);
    }

    // acc[j]: lanes 0-15 hold pixel j, lanes 16-31 hold pixel j+8 (any column).
    float ev[8];
    float lmin = 3.4e38f, lmax = 0.f;
#pragma unroll
    for (int j = 0; j < 8; ++j) {
        ev[j] = acc[j] * (1.0f / (float)C_);
        lmin = fminf(lmin, ev[j]);
        lmax = fmaxf(lmax, ev[j]);
    }

    const int wbase = wt * 128 + wave * 16;
    float* orow = out + (size_t)b * HW_ + (size_t)h * W_;
    if (lane == 0) {
#pragma unroll
        for (int j = 0; j < 8; ++j) orow[wbase + j] = ev[j];
    } else if (lane == 16) {
#pragma unroll
        for (int j = 0; j < 8; ++j) orow[wbase + 8 + j] = ev[j];
    }

    // Wave reduce (halves are duplicates; xor-16 merges pixel sets 0-7 / 8-15)
#pragma unroll
    for (int off = 16; off >= 1; off >>= 1) {
        lmin = fminf(lmin, __shfl_xor(lmin, off, 32));
        lmax = fmaxf(lmax, __shfl_xor(lmax, off, 32));
    }
    if (lane == 0) { sMin[wave] = lmin; sMax[wave] = lmax; }
    __syncthreads();
    if (tid == 0) {
        float bmin = sMin[0], bmax = sMax[0];
#pragma unroll
        for (int k = 1; k < 8; ++k) {
            bmin = fminf(bmin, sMin[k]);
            bmax = fmaxf(bmax, sMax[k]);
        }
        // values are >= 0, so uint ordering == float ordering
        atomicMin(&mm[b],     __float_as_uint(bmin));
        atomicMax(&mm[4 + b], __float_as_uint(bmax));
    }
}

__global__ void __launch_bounds__(256)
entropy_normalize_kernel(float* __restrict__ out,
                         const unsigned* __restrict__ mm, int total) {
    int idx = blockIdx.x * blockDim.x + threadIdx.x;
    if (idx >= total) return;
    int b = idx / HW_;
    float emin = __uint_as_float(mm[b]);
    float emax = __uint_as_float(mm[4 + b]);
    float denom = fmaxf(emax - emin, 1e-6f);
    out[idx] = (out[idx] - emin) * __builtin_amdgcn_rcpf(denom);
}

extern "C" void kernel_launch(void* const* d_in, const int* in_sizes, int n_in,
                              void* d_out, int out_size, void* d_ws, size_t ws_size,
                              hipStream_t stream) {
    (void)in_sizes; (void)n_in; (void)ws_size;
    const float* x = (const float*)d_in[0];
    float* out = (float*)d_out;
    unsigned* mm = (unsigned*)d_ws;   // [0..3] = per-batch min bits, [4..7] = max bits

    hipLaunchKernelGGL(entropy_init_minmax, dim3(1), dim3(8), 0, stream, mm);

    const int nblocks = B_ * H_ * (W_ / 128);   // 2048
    hipLaunchKernelGGL(entropy_map_kernel, dim3(nblocks), dim3(256), 0, stream,
                       x, out, mm);

    const int total = B_ * HW_;
    hipLaunchKernelGGL(entropy_normalize_kernel, dim3((total + 255) / 256),
                       dim3(256), 0, stream, out, mm, total);
}